// GResNet_20607253086494
// MI455X (gfx1250) — compile-verified
//
#include <hip/hip_runtime.h>

// ---------------------------------------------------------------------------
// GCN (DGL GraphConv, norm='both') pipeline for MI455X / gfx1250.
// Aggregations are HBM/atomic bound (~4 GB traffic -> ~170us @ 23.3 TB/s);
// GEMMs (~5.7 GFLOP) are fused WMMA f32 16x16x4 tiles staged through LDS.
// ---------------------------------------------------------------------------

#define GCN_N 50000
#define GCN_E 800000
#define GCN_D 128

typedef __attribute__((ext_vector_type(2))) float v2f;
typedef __attribute__((ext_vector_type(8))) float v8f;

// Native no-return f32 global atomic add (avoids CAS loop). Tracked by
// STOREcnt; S_ENDPGM performs an implicit wait-idle, and inter-kernel
// stream ordering makes results visible to the next kernel.
__device__ __forceinline__ void atomic_add_f32(float* addr, float val) {
    asm volatile("global_atomic_add_f32 %0, %1, off" :: "v"(addr), "v"(val) : "memory");
}

// ---------------------------------------------------------------------------
// Degrees: one thread per edge, two native f32 atomics.
// ---------------------------------------------------------------------------
__global__ void degree_kernel(const int* __restrict__ src, const int* __restrict__ dst,
                              float* __restrict__ indeg, float* __restrict__ outdeg, int nE) {
    int e = blockIdx.x * blockDim.x + threadIdx.x;
    if (e < nE) {
        atomic_add_f32(indeg + dst[e], 1.0f);
        atomic_add_f32(outdeg + src[e], 1.0f);
    }
}

__global__ void norm_kernel(const float* __restrict__ indeg, const float* __restrict__ outdeg,
                            float* __restrict__ innorm, float* __restrict__ outnorm, int n) {
    int i = blockIdx.x * blockDim.x + threadIdx.x;
    if (i < n) {
        innorm[i]  = rsqrtf(fmaxf(indeg[i], 1.0f));
        outnorm[i] = rsqrtf(fmaxf(outdeg[i], 1.0f));
    }
}

// ---------------------------------------------------------------------------
// Edge aggregation: out[dst] += h[src] * scale[src].
// One wave32 per edge; each lane owns 4 contiguous floats (global_load_b128
// -> 4 native f32 atomics). This is the bandwidth-critical kernel.
// ---------------------------------------------------------------------------
__global__ void aggregate_kernel(const float* __restrict__ h, const float* __restrict__ scale,
                                 const int* __restrict__ src, const int* __restrict__ dst,
                                 float* __restrict__ out, int nE) {
    int e    = (blockIdx.x * blockDim.x + threadIdx.x) >> 5;
    int lane = threadIdx.x & 31;
    if (e >= nE) return;
    int s = src[e];
    int d = dst[e];
    float sc = scale[s];
    const float4* hv = reinterpret_cast<const float4*>(h);
    float4 v = hv[(size_t)s * (GCN_D / 4) + lane];     // global_load_b128
    float* o = out + (size_t)d * GCN_D + lane * 4;
    atomic_add_f32(o + 0, v.x * sc);
    atomic_add_f32(o + 1, v.y * sc);
    atomic_add_f32(o + 2, v.z * sc);
    atomic_add_f32(o + 3, v.w * sc);
}

// ---------------------------------------------------------------------------
// Fused GEMM: out = RELU?( (innorm .* X) @ W + b [+ innorm .* res] )
//   X: N x 128 (aggregated features), W: 128 x DOUT, out: N x DOUT.
// grid.x = N/16 row tiles; block = (DOUT/16) waves; wave w -> 16-col tile w.
// The 16x128 X tile is staged in LDS (scaled by innorm at fill, padded row
// stride to spread banks). K-loop uses v_wmma_f32_16x16x4_f32. DOUT/RELU/RES
// are template params so all strides are immediates (B loads become
// global_load_b32 with 24-bit immediate offsets, epilogue is branch-free).
//
// f32 WMMA fragment layout (wave32):
//   A (16x4):  lane = half*16 + m ; comps hold K = half*2, half*2+1
//   B (4x16):  lane = half*16 + n ; comps hold K = half*2, half*2+1
//   C/D:       acc[r] -> M = r + half*8, N = lane&15
// ---------------------------------------------------------------------------
#define TILE_STRIDE 136   // 128 + pad; keeps float2/float4 alignment, spreads banks

template <int DOUT, bool RELU, bool HAS_RES>
__global__ void gemm_kernel(const float* __restrict__ X, const float* __restrict__ W,
                            const float* __restrict__ bias, const float* __restrict__ innorm,
                            const float* __restrict__ res, float* __restrict__ out) {
    __shared__ float tileX[16 * TILE_STRIDE];

    const int row0 = blockIdx.x * 16;

    // --- cooperative fill: tileX in LDS holds X rows row0..row0+15, each scaled by its innorm
    const float4* Xv = reinterpret_cast<const float4*>(X);
    for (int i = threadIdx.x; i < 16 * (GCN_D / 4); i += blockDim.x) {
        int row = i >> 5;          // 32 float4 per row
        int c4  = i & 31;
        float s = innorm[row0 + row];
        float4 v = Xv[(size_t)(row0 + row) * (GCN_D / 4) + c4];
        v.x *= s; v.y *= s; v.z *= s; v.w *= s;
        reinterpret_cast<float4*>(tileX + row * TILE_STRIDE)[c4] = v;   // ds_store_b128
    }
    __syncthreads();

    const int wave = threadIdx.x >> 5;
    const int lane = threadIdx.x & 31;
    const int half = lane >> 4;        // 0 | 1
    const int mn   = lane & 15;        // M for A-frag, N for B/D-frag
    const int ksel = half * 2;
    const int col0 = wave * 16;

    // Fixed base pointers; per-iteration offsets are compile-time immediates.
    const float* Wp0 = W + (size_t)ksel * DOUT + col0 + mn;        // K = k0+ksel
    const float* Wp1 = Wp0 + DOUT;                                  // K = k0+ksel+1
    const float* lp  = tileX + mn * TILE_STRIDE + ksel;

    v8f acc = {};
    #pragma unroll
    for (int k0 = 0; k0 < GCN_D; k0 += 4) {
        v2f a = *reinterpret_cast<const v2f*>(lp + k0);             // ds_load_b64
        v2f b;
        b.x = Wp0[(size_t)k0 * DOUT];                               // imm-offset b32
        b.y = Wp1[(size_t)k0 * DOUT];
        acc = __builtin_amdgcn_wmma_f32_16x16x4_f32(
            /*neg_a=*/false, a, /*neg_b=*/false, b,
            /*c_mod=*/(short)0, acc, /*reuse_a=*/false, /*reuse_b=*/false);
    }

    // --- epilogue: bias (+ residual * innorm) (+ relu); branch-free
    const float bv = bias[col0 + mn];
    const int rbase = row0 + half * 8;
    #pragma unroll
    for (int r = 0; r < 8; ++r) {
        const int row = rbase + r;
        float v = acc[r] + bv;
        if constexpr (HAS_RES) v += res[(size_t)row * GCN_D + col0 + mn] * innorm[row];
        if constexpr (RELU)    v = fmaxf(v, 0.0f);
        out[(size_t)row * DOUT + col0 + mn] = v;
    }
}

// ---------------------------------------------------------------------------
// Host-side orchestration (graph-capture safe: async memsets + kernels only).
// ---------------------------------------------------------------------------
extern "C" void kernel_launch(void* const* d_in, const int* in_sizes, int n_in,
                              void* d_out, int out_size, void* d_ws, size_t ws_size,
                              hipStream_t stream) {
    (void)in_sizes; (void)n_in; (void)out_size; (void)ws_size;

    const float* features = (const float*)d_in[0];
    const int*   src      = (const int*)d_in[1];
    const int*   dst      = (const int*)d_in[2];
    const float* W0 = (const float*)d_in[3];  const float* b0 = (const float*)d_in[4];
    const float* W1 = (const float*)d_in[5];  const float* b1 = (const float*)d_in[6];
    const float* W2 = (const float*)d_in[7];  const float* b2 = (const float*)d_in[8];
    const float* W3 = (const float*)d_in[9];  const float* b3 = (const float*)d_in[10];
    float* out = (float*)d_out;

    const int N_ = GCN_N, E_ = GCN_E;
    const size_t ND = (size_t)N_ * GCN_D;

    float* wsf     = (float*)d_ws;
    float* indeg   = wsf;                 // N
    float* outdeg  = indeg + N_;          // N
    float* innorm  = outdeg + N_;         // N
    float* outnorm = innorm + N_;         // N
    float* agg     = outnorm + N_;        // N*128
    float* hbuf    = agg + ND;            // N*128
    float* resbuf  = hbuf + ND;           // N*128

    const dim3 blk256(256);
    const int edgeBlocks = (E_ * 32 + 255) / 256;   // 1 wave per edge
    const int rowTiles   = N_ / 16;                 // 3125, exact

    // degrees + norms
    hipMemsetAsync(indeg, 0, 2 * (size_t)N_ * sizeof(float), stream);
    degree_kernel<<<(E_ + 255) / 256, blk256, 0, stream>>>(src, dst, indeg, outdeg, E_);
    norm_kernel<<<(N_ + 255) / 256, blk256, 0, stream>>>(indeg, outdeg, innorm, outnorm, N_);

    // layer-0 aggregate + shared residual aggregate (features are constant)
    hipMemsetAsync(agg, 0, ND * sizeof(float), stream);
    hipMemsetAsync(resbuf, 0, ND * sizeof(float), stream);
    aggregate_kernel<<<edgeBlocks, blk256, 0, stream>>>(features, outnorm, src, dst, agg, E_);
    aggregate_kernel<<<edgeBlocks, blk256, 0, stream>>>(features, innorm, src, dst, resbuf, E_);

    // layer 0: h = relu(innorm.*agg @ W0 + b0)
    gemm_kernel<128, true, false><<<rowTiles, 256, 0, stream>>>(agg, W0, b0, innorm, nullptr, hbuf);

    // layer 1: h = relu(innorm.*agg(h*outnorm) @ W1 + b1 + innorm.*resbuf)
    hipMemsetAsync(agg, 0, ND * sizeof(float), stream);
    aggregate_kernel<<<edgeBlocks, blk256, 0, stream>>>(hbuf, outnorm, src, dst, agg, E_);
    gemm_kernel<128, true, true><<<rowTiles, 256, 0, stream>>>(agg, W1, b1, innorm, resbuf, hbuf);

    // layer 2
    hipMemsetAsync(agg, 0, ND * sizeof(float), stream);
    aggregate_kernel<<<edgeBlocks, blk256, 0, stream>>>(hbuf, outnorm, src, dst, agg, E_);
    gemm_kernel<128, true, true><<<rowTiles, 256, 0, stream>>>(agg, W2, b2, innorm, resbuf, hbuf);

    // output layer: d_out = innorm.*agg(h*outnorm) @ W3 + b3   (DOUT=64, no relu)
    hipMemsetAsync(agg, 0, ND * sizeof(float), stream);
    aggregate_kernel<<<edgeBlocks, blk256, 0, stream>>>(hbuf, outnorm, src, dst, agg, E_);
    gemm_kernel<64, false, false><<<rowTiles, 128, 0, stream>>>(agg, W3, b3, innorm, nullptr, out);
}